// StateStack_32581621908206
// MI455X (gfx1250) — compile-verified
//
#include <hip/hip_runtime.h>

// StateStack gather: out[r, b, :] = hidden_stack[pos[b]-1+r, b, :], r in {0,1}.
// Pure bandwidth kernel: ~64 MB moved -> ~2.8 us at 23.3 TB/s on MI455X.
// Inputs (setup_inputs order): 0=x (unused), 1=hidden_stack, 2=op (unused), 3=pos.
//
// pos in [1,64] => scattered row pos+1 never overlaps gathered rows pos-1/pos
// within a batch column, and columns are independent => output depends only on
// the original hidden_stack. x and op are dead for the output.

constexpr int kBatch  = 8192;
constexpr int kHidden = 512;   // floats per row -> 2048 bytes -> 128 x 16B chunks

#define GLOBAL_AS __attribute__((address_space(1)))
#define LDS_AS    __attribute__((address_space(3)))

typedef int   v4i __attribute__((ext_vector_type(4)));
typedef float v4f __attribute__((ext_vector_type(4)));

#if defined(__HIP_DEVICE_COMPILE__) &&                                     \
    __has_builtin(__builtin_amdgcn_global_load_async_to_lds_b128) &&       \
    __has_builtin(__builtin_amdgcn_global_store_async_from_lds_b128)
#define USE_ASYNC_LDS 1
#else
#define USE_ASYNC_LDS 0
#endif

__device__ __forceinline__ void wait_async_zero() {
#if defined(__HIP_DEVICE_COMPILE__) && __has_builtin(__builtin_amdgcn_s_wait_asynccnt)
  __builtin_amdgcn_s_wait_asynccnt(0);
#elif defined(__HIP_DEVICE_COMPILE__)
  asm volatile("s_wait_asynccnt 0" ::: "memory");
#endif
}

__global__ __launch_bounds__(256) void state_stack_gather(
    const float* __restrict__ stack,  // (SEQ_LEN+2, BATCH, HIDDEN)
    const int*   __restrict__ pos,    // (BATCH,)
    float*       __restrict__ out)    // (2, BATCH, HIDDEN)
{
  const int b = blockIdx.x;          // batch column
  const int t = threadIdx.x;         // 0..255
  const int r = t >> 7;              // 0 or 1 (which gathered row)
  const int c = t & 127;             // 16B chunk within the 512-float row

  const int p = pos[b];              // block-uniform -> scalar load

  // Source row: pos-1 (r=0) or pos (r=1). Each (row,b) slice is 2048
  // contiguous bytes -> fully coalesced 16B/lane accesses.
  const float* src = stack + ((size_t)(p - 1 + r) * kBatch + b) * kHidden + (c << 2);
  float*       dst = out   + ((size_t)r * kBatch + b) * kHidden + (c << 2);

#if USE_ASYNC_LDS
  // gfx1250 async data-mover path: global -> LDS -> global, 16B per lane,
  // no VGPR staging. Each lane stores exactly the bytes it loaded, so a
  // wave-level s_wait_asynccnt between the two phases is sufficient
  // (no cross-lane sharing, no workgroup barrier needed).
  // Builtin signature (probe-confirmed): (AS1 v4i*, AS3 v4i*, imm, imm).
  __shared__ float lds_buf[256 * 4];   // 4 KB: one 16B slot per thread

  GLOBAL_AS v4i* gsrc  = (GLOBAL_AS v4i*)src;   // casts away const + addrspace
  GLOBAL_AS v4i* gdst  = (GLOBAL_AS v4i*)dst;
  LDS_AS    v4i* lslot = (LDS_AS    v4i*)&lds_buf[t << 2];

  __builtin_amdgcn_global_load_async_to_lds_b128(gsrc, lslot, 0, 0);
  wait_async_zero();   // async loads complete (in order) before stores read LDS
  __builtin_amdgcn_global_store_async_from_lds_b128(gdst, lslot, 0, 0);
  wait_async_zero();   // drain before wave exit
#else
  // Fallback: single 128-bit non-temporal load + store per thread.
  // Every byte is touched exactly once -> NT hints avoid polluting WGP$/L2.
  // Use clang ext-vector (not HIP float4 class) so the builtin accepts it.
  v4f v = __builtin_nontemporal_load((const v4f*)src);
  __builtin_nontemporal_store(v, (v4f*)dst);
#endif
}

extern "C" void kernel_launch(void* const* d_in, const int* in_sizes, int n_in,
                              void* d_out, int out_size, void* d_ws, size_t ws_size,
                              hipStream_t stream) {
  (void)in_sizes; (void)n_in; (void)out_size; (void)d_ws; (void)ws_size;
  const float* stack = (const float*)d_in[1];   // hidden_stack
  const int*   pos   = (const int*)d_in[3];     // pos
  float*       out   = (float*)d_out;

  // One block per batch column; 256 threads cover both gathered rows
  // (2 rows x 128 x 16B chunks). 8192 blocks x 8 waves saturates HBM.
  state_stack_gather<<<dim3(kBatch), dim3(256), 0, stream>>>(stack, pos, out);
}